// EntityClassify_54305566491326
// MI455X (gfx1250) — compile-verified
//
#include <hip/hip_runtime.h>

typedef __attribute__((ext_vector_type(2))) float v2f;
typedef __attribute__((ext_vector_type(8))) float v8f;

#define HIN 128

// ---------------------------------------------------------------------------
// Y[nrows, NOUT] = relu?(X[nrows,128]) @ W[128, NOUT]  via V_WMMA_F32_16X16X4_F32
// Block: 256 threads = 8 waves; block computes 128 rows x NOUT cols.
// K consumed in 32-wide chunks staged in LDS (X row-major, W transposed),
// LDS rows padded to 34 floats -> conflict-free ds_load_b64 fragment reads.
// ---------------------------------------------------------------------------
template <int NOUT>
__global__ __launch_bounds__(256) void gemm_wmma_f32(
    const float* __restrict__ X, const float* __restrict__ W,
    float* __restrict__ Y, int nrows, int relu_in) {
  constexpr int KC  = 32;       // K chunk
  constexpr int LDX = KC + 2;   // 34-float pitch: bank-conflict-free b64 reads
  constexpr int NT  = NOUT / 16;

  __shared__ float Xs[128 * LDX];
  __shared__ float Ws[NOUT * LDX];

  const int tid   = threadIdx.x;
  const int wave  = tid >> 5;
  const int lane  = tid & 31;
  const int mbase = wave * 16;                 // wave's 16-row tile in block
  const long rowblock = (long)blockIdx.x * 128;

  v8f acc[NT] = {};

  for (int kc = 0; kc < HIN; kc += KC) {
    // ---- stage X tile (128 rows x 32 cols), fused input ReLU ----
    {
      const int r0 = tid >> 3;            // 0..31
      const int c4 = (tid & 7) << 2;      // 0,4,...,28
#pragma unroll
      for (int rr = 0; rr < 4; ++rr) {
        const int  row  = r0 + rr * 32;
        const long grow = rowblock + row;
        float4 v = make_float4(0.f, 0.f, 0.f, 0.f);
        if (grow < nrows) v = *(const float4*)(X + grow * HIN + kc + c4);
        if (relu_in) {
          v.x = fmaxf(v.x, 0.f); v.y = fmaxf(v.y, 0.f);
          v.z = fmaxf(v.z, 0.f); v.w = fmaxf(v.w, 0.f);
        }
        float* p = &Xs[row * LDX + c4];
        p[0] = v.x; p[1] = v.y; p[2] = v.z; p[3] = v.w;
      }
    }
    // ---- stage W tile transposed: Ws[n][k] = W[kc+k][n] (coalesced reads) ----
    for (int i = tid; i < KC * NOUT; i += 256) {
      const int k = i / NOUT;
      const int n = i - k * NOUT;
      Ws[n * LDX + k] = W[(kc + k) * NOUT + n];
    }
    __syncthreads();

#pragma unroll
    for (int kk = 0; kk < KC; kk += 4) {
      // A frag (16x4 f32): lanes 0-15 -> K=kk,kk+1 ; lanes 16-31 -> K=kk+2,kk+3
      const int ks  = kk + ((lane >> 4) << 1);
      const int l16 = lane & 15;
      const v2f a = *(const v2f*)(&Xs[(mbase + l16) * LDX + ks]);
#pragma unroll
      for (int nt = 0; nt < NT; ++nt) {
        const v2f b = *(const v2f*)(&Ws[(nt * 16 + l16) * LDX + ks]);
        acc[nt] = __builtin_amdgcn_wmma_f32_16x16x4_f32(
            false, a, false, b, (short)0, acc[nt], false, false);
      }
    }
    __syncthreads();
  }

  // C/D layout: VGPR r -> lanes 0-15: (M=r, N=lane); lanes 16-31: (M=r+8)
  const int n  = lane & 15;
  const int mo = (lane >> 4) * 8;
#pragma unroll
  for (int nt = 0; nt < NT; ++nt) {
#pragma unroll
    for (int r = 0; r < 8; ++r) {
      const long grow = rowblock + mbase + mo + r;
      if (grow < nrows) Y[grow * NOUT + nt * 16 + n] = acc[nt][r];
    }
  }
}

// ---------------------------------------------------------------------------
// CSR construction (once per launch, reused by all 3 layers):
//   zero -> histogram(dst) -> chunk sums -> single-block scan -> chunk scan
//   -> fill csr_src via per-row cursors.
// ---------------------------------------------------------------------------
__global__ void zero_i32(int* __restrict__ p, int n) {
  for (int i = blockIdx.x * 256 + threadIdx.x; i < n; i += gridDim.x * 256)
    p[i] = 0;
}

__global__ void hist_kernel(const int* __restrict__ dst, int* __restrict__ deg,
                            int ne) {
  for (int i = blockIdx.x * 256 + threadIdx.x; i < ne; i += gridDim.x * 256)
    atomicAdd(&deg[dst[i]], 1);
}

// block b sums deg[b*2048 .. b*2048+2047] -> bsum[b]
__global__ __launch_bounds__(256) void chunk_sum(const int* __restrict__ deg,
                                                 int* __restrict__ bsum, int n) {
  __shared__ int s[256];
  const int base = blockIdx.x * 2048;
  const int t = threadIdx.x;
  int sum = 0;
#pragma unroll
  for (int j = 0; j < 8; ++j) {
    const int idx = base + t * 8 + j;
    if (idx < n) sum += deg[idx];
  }
  s[t] = sum;
  __syncthreads();
  for (int off = 128; off > 0; off >>= 1) {
    if (t < off) s[t] += s[t + off];
    __syncthreads();
  }
  if (t == 0) bsum[blockIdx.x] = s[0];
}

// single block: exclusive scan of bsum[0..nb) in place (nb <= 256); total -> *total
__global__ __launch_bounds__(256) void scan_small(int* __restrict__ bsum, int nb,
                                                  int* __restrict__ total) {
  __shared__ int s[256];
  const int t = threadIdx.x;
  const int v = (t < nb) ? bsum[t] : 0;
  s[t] = v;
  __syncthreads();
  for (int off = 1; off < 256; off <<= 1) {
    const int x = (t >= off) ? s[t - off] : 0;
    __syncthreads();
    s[t] += x;
    __syncthreads();
  }
  if (t < nb) bsum[t] = s[t] - v;  // exclusive
  if (t == nb - 1) *total = s[t];  // inclusive grand total -> rowstart[n]
}

// block b: exclusive scan of its 2048 degrees + bsum[b]; writes rowstart & cursor
__global__ __launch_bounds__(256) void chunk_scan(const int* __restrict__ deg,
                                                  const int* __restrict__ bsum,
                                                  int* __restrict__ rowstart,
                                                  int* __restrict__ cursor, int n) {
  __shared__ int s[256];
  const int base = blockIdx.x * 2048;
  const int t = threadIdx.x;
  int loc[8];
  int sum = 0;
#pragma unroll
  for (int j = 0; j < 8; ++j) {
    const int idx = base + t * 8 + j;
    const int v = (idx < n) ? deg[idx] : 0;
    loc[j] = sum;
    sum += v;
  }
  s[t] = sum;
  __syncthreads();
  const int mysum = sum;
  for (int off = 1; off < 256; off <<= 1) {
    const int x = (t >= off) ? s[t - off] : 0;
    __syncthreads();
    s[t] += x;
    __syncthreads();
  }
  const int prefix = s[t] - mysum + bsum[blockIdx.x];
#pragma unroll
  for (int j = 0; j < 8; ++j) {
    const int idx = base + t * 8 + j;
    if (idx < n) {
      const int r = prefix + loc[j];
      rowstart[idx] = r;
      cursor[idx] = r;
    }
  }
}

__global__ void fill_kernel(const int* __restrict__ src, const int* __restrict__ dst,
                            int* __restrict__ cursor, int* __restrict__ csr_src,
                            int ne) {
  for (int i = blockIdx.x * 256 + threadIdx.x; i < ne; i += gridDim.x * 256) {
    const int pos = atomicAdd(&cursor[dst[i]], 1);
    csr_src[pos] = src[i];
  }
}

// ---------------------------------------------------------------------------
// Segment gather-accumulate: one wave per destination row (exclusive owner ->
// NO atomics). Lanes hold float4/float2 register accumulators; each edge is a
// coalesced 512B/256B row gather. out[row] += sum(Y[csr_src[e]]).
// ---------------------------------------------------------------------------
template <int FPL>  // floats per lane: 4 -> 128 cols, 2 -> 64 cols
__global__ __launch_bounds__(256) void seg_gather(
    const float* __restrict__ Y, const int* __restrict__ rowstart,
    const int* __restrict__ csr_src, float* __restrict__ out, int nrows) {
  constexpr int NC = FPL * 32;
  const int row  = (int)((blockIdx.x * 256 + threadIdx.x) >> 5);
  const int lane = threadIdx.x & 31;
  if (row >= nrows) return;
  const int e0 = rowstart[row];
  const int e1 = rowstart[row + 1];
  if constexpr (FPL == 4) {
    float4 acc = make_float4(0.f, 0.f, 0.f, 0.f);
    for (int e = e0; e < e1; ++e) {
      const int s = csr_src[e];
      const float4 v = *(const float4*)(Y + (size_t)s * NC + lane * 4);
      acc.x += v.x; acc.y += v.y; acc.z += v.z; acc.w += v.w;
    }
    float* op = out + (size_t)row * NC + lane * 4;
    float4 cur = *(float4*)op;
    cur.x += acc.x; cur.y += acc.y; cur.z += acc.z; cur.w += acc.w;
    *(float4*)op = cur;
  } else {
    float2 acc = make_float2(0.f, 0.f);
    for (int e = e0; e < e1; ++e) {
      const int s = csr_src[e];
      const float2 v = *(const float2*)(Y + (size_t)s * NC + lane * 2);
      acc.x += v.x; acc.y += v.y;
    }
    float* op = out + (size_t)row * NC + lane * 2;
    float2 cur = *(float2*)op;
    cur.x += acc.x; cur.y += acc.y;
    *(float2*)op = cur;
  }
}

__global__ void zero4_kernel(float* __restrict__ p, size_t n) {
  const size_t n4 = n >> 2;
  const float4 z = make_float4(0.f, 0.f, 0.f, 0.f);
  for (size_t i = (size_t)blockIdx.x * blockDim.x + threadIdx.x; i < n4;
       i += (size_t)gridDim.x * blockDim.x)
    ((float4*)p)[i] = z;
}

__global__ void relu4_kernel(float* __restrict__ p, size_t n) {
  const size_t n4 = n >> 2;
  for (size_t i = (size_t)blockIdx.x * blockDim.x + threadIdx.x; i < n4;
       i += (size_t)gridDim.x * blockDim.x) {
    float4 v = ((float4*)p)[i];
    v.x = fmaxf(v.x, 0.f); v.y = fmaxf(v.y, 0.f);
    v.z = fmaxf(v.z, 0.f); v.w = fmaxf(v.w, 0.f);
    ((float4*)p)[i] = v;
  }
}

extern "C" void kernel_launch(void* const* d_in, const int* in_sizes, int n_in,
                              void* d_out, int out_size, void* d_ws, size_t ws_size,
                              hipStream_t stream) {
  (void)n_in; (void)ws_size;
  const float* x_paper  = (const float*)d_in[0];
  const float* x_author = (const float*)d_in[1];
  const float* W_hidden = (const float*)d_in[3];  // [2][4][128][128]
  const float* W_out    = (const float*)d_in[4];  // [4][128][64]
  const int* src_w  = (const int*)d_in[5];
  const int* dst_w  = (const int*)d_in[6];
  const int* src_wb = (const int*)d_in[7];
  const int* dst_wb = (const int*)d_in[8];
  const int* src_c  = (const int*)d_in[9];
  const int* dst_c  = (const int*)d_in[10];
  const int* src_t  = (const int*)d_in[11];
  const int* dst_t  = (const int*)d_in[12];
  const int EW = in_sizes[5], EWB = in_sizes[7], EC = in_sizes[9], ET = in_sizes[11];

  const int NP = 200000, NA = 100000, NF = 60000;
  const size_t H_ELEMS = (size_t)(NP + NA + NF) * 128;

  // ---- workspace layout ----
  float* h0 = (float*)d_ws;
  float* h1 = h0 + H_ELEMS;
  float* yb = h1 + H_ELEMS;  // 200000*128 floats (per-relation GEMM output)
  int* ib    = (int*)(yb + (size_t)NP * 128);
  int* rs_w  = ib;                 int* rs_wb = rs_w + (NP + 1);
  int* rs_c  = rs_wb + (NA + 1);   int* rs_t  = rs_c + (NP + 1);
  int* cu_w  = rs_t + (NF + 1);    int* cu_wb = cu_w + NP;
  int* cu_c  = cu_wb + NA;         int* cu_t  = cu_c + NP;
  int* cs_w  = cu_t + NF;          int* cs_wb = cs_w + EW;
  int* cs_c  = cs_wb + EWB;        int* cs_t  = cs_c + EC;
  int* bsum  = cs_t + ET;          // 256 ints scratch

  auto zerof = [&](float* p, size_t nelem) {
    int blocks = (int)((nelem / 4 + 255) / 256);
    if (blocks > 4096) blocks = 4096;
    zero4_kernel<<<blocks, 256, 0, stream>>>(p, nelem);
  };
  auto gsz = [](int n) { int b = (n + 255) / 256; return b > 4096 ? 4096 : b; };

  // ---- build CSR-by-destination for each relation (reused for all 3 layers) ----
  auto build_csr = [&](const int* src, const int* dst, int ne, int n,
                       int* rowstart, int* cursor, int* csr) {
    const int nb = (n + 2047) / 2048;  // <= 98 for n=200000 (fits scan_small)
    zero_i32<<<gsz(n), 256, 0, stream>>>(cursor, n);
    hist_kernel<<<gsz(ne), 256, 0, stream>>>(dst, cursor, ne);
    chunk_sum<<<nb, 256, 0, stream>>>(cursor, bsum, n);
    scan_small<<<1, 256, 0, stream>>>(bsum, nb, rowstart + n);
    chunk_scan<<<nb, 256, 0, stream>>>(cursor, bsum, rowstart, cursor, n);
    fill_kernel<<<gsz(ne), 256, 0, stream>>>(src, dst, cursor, csr, ne);
  };
  build_csr(src_w,  dst_w,  EW,  NP, rs_w,  cu_w,  cs_w);
  build_csr(src_wb, dst_wb, EWB, NA, rs_wb, cu_wb, cs_wb);
  build_csr(src_c,  dst_c,  EC,  NP, rs_c,  cu_c,  cs_c);
  build_csr(src_t,  dst_t,  ET,  NF, rs_t,  cu_t,  cs_t);

  auto gather128 = [&](const int* rs, const int* cs, float* o, int n) {
    seg_gather<4><<<(n * 32 + 255) / 256, 256, 0, stream>>>(yb, rs, cs, o, n);
  };
  auto gather64 = [&](const int* rs, const int* cs, float* o, int n) {
    seg_gather<2><<<(n * 32 + 255) / 256, 256, 0, stream>>>(yb, rs, cs, o, n);
  };

  const float* xp = x_paper;
  const float* xa = x_author;

  // ---- two hidden layers (H=128) ----
  for (int l = 0; l < 2; ++l) {
    float* hout = (l == 0) ? h0 : h1;
    float* hp = hout;
    float* ha = hout + (size_t)NP * 128;
    float* hf = ha + (size_t)NA * 128;
    const float* Wl = W_hidden + (size_t)l * 4 * 128 * 128;
    const int relu_in = (l > 0);
    zerof(hout, H_ELEMS);

    // rel 0: author -> paper (writes)
    gemm_wmma_f32<128><<<(NA + 127) / 128, 256, 0, stream>>>(xa, Wl + 0 * 16384, yb, NA, relu_in);
    gather128(rs_w, cs_w, hp, NP);
    // rel 1: paper -> author (written_by)
    gemm_wmma_f32<128><<<(NP + 127) / 128, 256, 0, stream>>>(xp, Wl + 1 * 16384, yb, NP, relu_in);
    gather128(rs_wb, cs_wb, ha, NA);
    // rel 2: paper -> paper (cites)
    gemm_wmma_f32<128><<<(NP + 127) / 128, 256, 0, stream>>>(xp, Wl + 2 * 16384, yb, NP, relu_in);
    gather128(rs_c, cs_c, hp, NP);
    // rel 3: paper -> field (has_topic)
    gemm_wmma_f32<128><<<(NP + 127) / 128, 256, 0, stream>>>(xp, Wl + 3 * 16384, yb, NP, relu_in);
    gather128(rs_t, cs_t, hf, NF);

    xp = hp;
    xa = ha;
  }

  // ---- output layer (NOUT=64) straight into d_out, then final ReLU ----
  float* out = (float*)d_out;
  float* op = out;
  float* oa = out + (size_t)NP * 64;
  float* of = oa + (size_t)NA * 64;
  zerof(out, (size_t)out_size);

  gemm_wmma_f32<64><<<(NA + 127) / 128, 256, 0, stream>>>(xa, W_out + 0 * 8192, yb, NA, 1);
  gather64(rs_w, cs_w, op, NP);

  gemm_wmma_f32<64><<<(NP + 127) / 128, 256, 0, stream>>>(xp, W_out + 1 * 8192, yb, NP, 1);
  gather64(rs_wb, cs_wb, oa, NA);

  gemm_wmma_f32<64><<<(NP + 127) / 128, 256, 0, stream>>>(xp, W_out + 2 * 8192, yb, NP, 1);
  gather64(rs_c, cs_c, op, NP);

  gemm_wmma_f32<64><<<(NP + 127) / 128, 256, 0, stream>>>(xp, W_out + 3 * 8192, yb, NP, 1);
  gather64(rs_t, cs_t, of, NF);

  relu4_kernel<<<4096, 256, 0, stream>>>(out, (size_t)out_size);
}